// ILRMA_V2_6803228197168
// MI455X (gfx1250) — compile-verified
//
#include <hip/hip_runtime.h>
#include <hip/hip_bf16.h>

// ---------------- problem constants ----------------
#define I_DIM   2049
#define J_DIM   3000
#define K_DIM   8
#define N_ITERS 30
#define NMF_EPS 1e-20f
#define IP_EPS  1e-20f

#define ITILES    129   // ceil(I/16)
#define JCHUNKS_T 6
#define JSPAN_T   500   // J / JCHUNKS_T, divisible by 4
#define JTILES_V  188   // ceil(J/16)
#define ICHUNKS_V 16

typedef float v2f __attribute__((ext_vector_type(2)));
typedef float v8f __attribute__((ext_vector_type(8)));

static __device__ __forceinline__ int imin(int a, int b) { return a < b ? a : b; }

// D(16x16,f32) += A(16x4,f32) x B(4x16,f32)  — full-precision CDNA5 WMMA
static __device__ __forceinline__ v8f wmma4(v2f a, v2f b, v8f c) {
  return __builtin_amdgcn_wmma_f32_16x16x4_f32(false, a, false, b, (short)0, c,
                                               false, false);
}

// ---------------- init kernels ----------------
__global__ __launch_bounds__(256) void ilrma_init_TVW(
    const float* __restrict__ Tin, const float* __restrict__ Vin,
    float* __restrict__ T0, float* __restrict__ T1,
    float* __restrict__ V0, float* __restrict__ V1, float* __restrict__ W) {
  int idx = blockIdx.x * 256 + threadIdx.x;
  if (idx < I_DIM * K_DIM) {  // T_init (I,K,M) -> (I,K) per source
    T0[idx] = Tin[2 * idx + 0];
    T1[idx] = Tin[2 * idx + 1];
  }
  if (idx < K_DIM * J_DIM) {  // V_init (K,J,M) -> (K,J) per source
    V0[idx] = Vin[2 * idx + 0];
    V1[idx] = Vin[2 * idx + 1];
  }
  if (idx < I_DIM) {  // W = I (complex 2x2 per bin, interleaved re/im)
    float* w = W + idx * 8;
    w[0] = 1.f; w[1] = 0.f; w[2] = 0.f; w[3] = 0.f;
    w[4] = 0.f; w[5] = 0.f; w[6] = 1.f; w[7] = 0.f;
  }
}

// ---------------- P = |W X|^2, layout (n, j, i) with i contiguous ----------------
__global__ __launch_bounds__(256) void ilrma_compute_P(
    const float* __restrict__ X, const float* __restrict__ W,
    float* __restrict__ P) {
  int idx = blockIdx.x * 256 + threadIdx.x;
  if (idx >= J_DIM * I_DIM) return;
  int j = idx / I_DIM, i = idx - j * I_DIM;
  const float* x0 = X + ((size_t)j * I_DIM + i) * 2;
  const float* x1 = X + ((size_t)(J_DIM + j) * I_DIM + i) * 2;
  float x0r = x0[0], x0i = x0[1], x1r = x1[0], x1i = x1[1];
  const float* w = W + i * 8;
#pragma unroll
  for (int n = 0; n < 2; ++n) {
    float wr0 = w[n * 4 + 0], wi0 = w[n * 4 + 1];
    float wr1 = w[n * 4 + 2], wi1 = w[n * 4 + 3];
    float yr = wr0 * x0r - wi0 * x0i + wr1 * x1r - wi1 * x1i;
    float yi = wr0 * x0i + wi0 * x0r + wr1 * x1i + wi1 * x1r;
    P[(size_t)n * J_DIM * I_DIM + idx] = yr * yr + yi * yi;
  }
}

// ---------------- T-update GEMM (contraction over j) via f32 WMMA ----------------
// partial[jc][i][k] : numer = sum_j YdR*V ; denom = sum_j Rd*V
__global__ __launch_bounds__(256) void ilrma_t_update(
    const float* __restrict__ Pn, const float* __restrict__ Told,
    const float* __restrict__ Vn, float* __restrict__ ptN,
    float* __restrict__ ptD) {
  const int wid = blockIdx.x * 8 + (threadIdx.x >> 5);
  if (wid >= ITILES * JCHUNKS_T) return;  // wave-uniform
  const int lane = threadIdx.x & 31;
  const int it = wid % ITILES, jc = wid / ITILES;
  const int i0 = it * 16;
  const int m = lane & 15;   // A row (i offset) / B col (k)
  const int half = lane >> 4;
  const int im = imin(i0 + m, I_DIM - 1);  // clamped A-row index

  float trow[K_DIM];
#pragma unroll
  for (int k = 0; k < K_DIM; ++k) trow[k] = Told[im * 8 + k];

  v8f accN = {0.f, 0.f, 0.f, 0.f, 0.f, 0.f, 0.f, 0.f};
  v8f accD = {0.f, 0.f, 0.f, 0.f, 0.f, 0.f, 0.f, 0.f};
  const int jbeg = jc * JSPAN_T;
  for (int js = 0; js < JSPAN_T; js += 4) {
    const int jA = jbeg + js + half * 2;  // this lane's two j's: jA, jA+1
    float p0 = Pn[(size_t)jA * I_DIM + im];
    float p1 = Pn[(size_t)(jA + 1) * I_DIM + im];
    float r0 = 0.f, r1 = 0.f, vb0 = 0.f, vb1 = 0.f;
#pragma unroll
    for (int k = 0; k < K_DIM; ++k) {
      float v0 = Vn[k * J_DIM + jA];
      float v1 = Vn[k * J_DIM + jA + 1];
      r0 = fmaf(trow[k], v0, r0);
      r1 = fmaf(trow[k], v1, r1);
      if (k == m) { vb0 = v0; vb1 = v1; }  // B element (zero for m>=8)
    }
    r0 = fabsf(r0); r1 = fabsf(r1);
    v2f a1, a2, b;
    a1.x = p0 / fmaf(r0, r0, NMF_EPS);  // YdR
    a1.y = p1 / fmaf(r1, r1, NMF_EPS);
    a2.x = 1.f / (r0 + NMF_EPS);        // Rd
    a2.y = 1.f / (r1 + NMF_EPS);
    b.x = vb0; b.y = vb1;
    accN = wmma4(a1, b, accN);
    accD = wmma4(a2, b, accD);
  }
  // D rows = i_rel (half*8 + r), cols = k (lane%16); only k<8 valid
  if (m < K_DIM) {
#pragma unroll
    for (int r = 0; r < 8; ++r) {
      int irow = i0 + half * 8 + r;
      if (irow < I_DIM) {
        ptN[(size_t)jc * I_DIM * 8 + irow * 8 + m] = accN[r];
        ptD[(size_t)jc * I_DIM * 8 + irow * 8 + m] = accD[r];
      }
    }
  }
}

__global__ __launch_bounds__(256) void ilrma_t_finalize(
    const float* __restrict__ Told, const float* __restrict__ ptN,
    const float* __restrict__ ptD, float* __restrict__ Tnew) {
  int idx = blockIdx.x * 256 + threadIdx.x;
  if (idx >= I_DIM * K_DIM) return;
  float sn = 0.f, sd = 0.f;
#pragma unroll
  for (int c = 0; c < JCHUNKS_T; ++c) {
    sn += ptN[(size_t)c * I_DIM * 8 + idx];
    sd += ptD[(size_t)c * I_DIM * 8 + idx];
  }
  Tnew[idx] = Told[idx] * sqrtf(sn / (sd + NMF_EPS));
}

// ---------------- V-update GEMM (contraction over i) via f32 WMMA ----------------
// partial[ic][k][j] : numer = sum_i Tnew*YdR ; denom = sum_i Tnew*Rd2
__global__ __launch_bounds__(256) void ilrma_v_accum(
    const float* __restrict__ Pn, const float* __restrict__ Told,
    const float* __restrict__ Tnew, const float* __restrict__ Vn,
    float* __restrict__ pvN, float* __restrict__ pvD) {
  __shared__ float tile[8][16][17];  // per-wave P tile [j_rel][i_rel]
  const int wslot = threadIdx.x >> 5;
  const int wid = blockIdx.x * 8 + wslot;
  const int lane = threadIdx.x & 31;
  const int jt = wid % JTILES_V, ic = wid / JTILES_V;
  const int j0 = jt * 16;
  const int m = lane & 15;   // A row (k) / B col (j_rel)
  const int half = lane >> 4;
  const int jm = imin(j0 + m, J_DIM - 1);  // this lane's j column

  float vcol[K_DIM];
#pragma unroll
  for (int k = 0; k < K_DIM; ++k) vcol[k] = Vn[k * J_DIM + jm];

  v8f accN = {0.f, 0.f, 0.f, 0.f, 0.f, 0.f, 0.f, 0.f};
  v8f accD = {0.f, 0.f, 0.f, 0.f, 0.f, 0.f, 0.f, 0.f};
  float(*tl)[17] = tile[wslot];
  const int t0 = (ic * ITILES) / ICHUNKS_V, t1 = ((ic + 1) * ITILES) / ICHUNKS_V;
  for (int tt = t0; tt < t1; ++tt) {
    const int ib = tt * 16;
    // cooperative (coalesced-in-i) load of P[j0..j0+15][ib..ib+15] into LDS
#pragma unroll
    for (int r = 0; r < 8; ++r) {
      int jj = imin(j0 + r * 2 + half, J_DIM - 1);
      int ii = imin(ib + m, I_DIM - 1);
      tl[r * 2 + half][m] = Pn[(size_t)jj * I_DIM + ii];
    }
    asm volatile("s_wait_dscnt 0" ::: "memory");  // LDS tile visible (in-wave)
#pragma unroll
    for (int s = 0; s < 4; ++s) {
      const int iA = ib + s * 4 + half * 2;  // this lane's two i's
      const int ic0 = imin(iA, I_DIM - 1), ic1 = imin(iA + 1, I_DIM - 1);
      float ro0 = 0.f, ro1 = 0.f, rn0 = 0.f, rn1 = 0.f;
      float tnm0 = 0.f, tnm1 = 0.f;
#pragma unroll
      for (int k = 0; k < K_DIM; ++k) {
        float to0 = Told[ic0 * 8 + k], to1 = Told[ic1 * 8 + k];
        float tn0 = Tnew[ic0 * 8 + k], tn1 = Tnew[ic1 * 8 + k];
        ro0 = fmaf(to0, vcol[k], ro0); ro1 = fmaf(to1, vcol[k], ro1);
        rn0 = fmaf(tn0, vcol[k], rn0); rn1 = fmaf(tn1, vcol[k], rn1);
        if (k == m) { tnm0 = tn0; tnm1 = tn1; }  // A element (zero for m>=8)
      }
      ro0 = fabsf(ro0); ro1 = fabsf(ro1);
      rn0 = fabsf(rn0); rn1 = fabsf(rn1);
      float p0 = tl[m][s * 4 + half * 2];
      float p1 = tl[m][s * 4 + half * 2 + 1];
      v2f a, bN, bD;
      a.x = (iA < I_DIM) ? tnm0 : 0.f;       // zero OOB i columns of A
      a.y = (iA + 1 < I_DIM) ? tnm1 : 0.f;
      bN.x = p0 / fmaf(ro0, ro0, NMF_EPS);   // YdR (from old T)
      bN.y = p1 / fmaf(ro1, ro1, NMF_EPS);
      bD.x = 1.f / (rn0 + NMF_EPS);          // Rd2 (from new T)
      bD.y = 1.f / (rn1 + NMF_EPS);
      accN = wmma4(a, bN, accN);
      accD = wmma4(a, bD, accD);
    }
  }
  // D rows = k (only 0..7, held by lanes 0..15), cols = j
  if (half == 0 && (j0 + m) < J_DIM) {
#pragma unroll
    for (int r = 0; r < 8; ++r) {
      pvN[((size_t)ic * 8 + r) * J_DIM + j0 + m] = accN[r];
      pvD[((size_t)ic * 8 + r) * J_DIM + j0 + m] = accD[r];
    }
  }
}

__global__ __launch_bounds__(256) void ilrma_v_finalize(
    const float* __restrict__ pvN, const float* __restrict__ pvD,
    float* __restrict__ V) {
  int idx = blockIdx.x * 256 + threadIdx.x;
  if (idx >= K_DIM * J_DIM) return;
  float sn = 0.f, sd = 0.f;
#pragma unroll
  for (int c = 0; c < ICHUNKS_V; ++c) {
    sn += pvN[(size_t)c * K_DIM * J_DIM + idx];
    sd += pvD[(size_t)c * K_DIM * J_DIM + idx];
  }
  V[idx] = V[idx] * sqrtf(sn / (sd + NMF_EPS));
}

// ---------------- per-bin weighted covariance + 2x2 complex IP solve ----------------
__global__ __launch_bounds__(256) void ilrma_dw_update(
    const float* __restrict__ X, const float* __restrict__ Tn,
    const float* __restrict__ Vn, float* __restrict__ W, int n) {
  __shared__ float Vs[K_DIM][256];
  const int i = blockIdx.x * 256 + threadIdx.x;
  const int ic = imin(i, I_DIM - 1);
  float t8[K_DIM];
#pragma unroll
  for (int k = 0; k < K_DIM; ++k) t8[k] = Tn[ic * 8 + k];

  float d00 = 0.f, d11 = 0.f, dre = 0.f, dim = 0.f;
  for (int jb = 0; jb < J_DIM; jb += 256) {
    int t = threadIdx.x;
#pragma unroll
    for (int k = 0; k < K_DIM; ++k)
      Vs[k][t] = (jb + t < J_DIM) ? Vn[k * J_DIM + jb + t] : 0.f;
    __syncthreads();
    const int jcount = imin(256, J_DIM - jb);
    for (int jj = 0; jj < jcount; ++jj) {
      float R = 0.f;
#pragma unroll
      for (int k = 0; k < K_DIM; ++k) R = fmaf(t8[k], Vs[k][jj], R);
      float inv = 1.f / (R + IP_EPS);
      const float* x0 = X + ((size_t)(jb + jj) * I_DIM + ic) * 2;
      const float* x1 = X + ((size_t)(J_DIM + jb + jj) * I_DIM + ic) * 2;
      float x0r = x0[0], x0i = x0[1], x1r = x1[0], x1i = x1[1];
      d00 = fmaf(x0r * x0r + x0i * x0i, inv, d00);
      d11 = fmaf(x1r * x1r + x1i * x1i, inv, d11);
      dre = fmaf(x0r * x1r + x0i * x1i, inv, dre);  // x0*conj(x1)
      dim = fmaf(x0i * x1r - x0r * x1i, inv, dim);
    }
    __syncthreads();
  }
  if (i >= I_DIM) return;
  const float sc = 1.f / (float)J_DIM;
  float D00 = d00 * sc + IP_EPS, D11 = d11 * sc + IP_EPS;  // D_reg
  float Dre = dre * sc, Dim = dim * sc;
  float* w = W + i * 8;
  float w00r = w[0], w00i = w[1], w01r = w[2], w01i = w[3];
  float w10r = w[4], w10i = w[5], w11r = w[6], w11i = w[7];
  // A = W * D_reg  (D10 = conj(D01))
  float a00r = w00r * D00 + w01r * Dre + w01i * Dim;
  float a00i = w00i * D00 + w01i * Dre - w01r * Dim;
  float a01r = w00r * Dre - w00i * Dim + w01r * D11;
  float a01i = w00i * Dre + w00r * Dim + w01i * D11;
  float a10r = w10r * D00 + w11r * Dre + w11i * Dim;
  float a10i = w10i * D00 + w11i * Dre - w11r * Dim;
  float a11r = w10r * Dre - w10i * Dim + w11r * D11;
  float a11i = w10i * Dre + w10r * Dim + w11i * D11;
  float detr = (a00r * a11r - a00i * a11i) - (a01r * a10r - a01i * a10i);
  float deti = (a00r * a11i + a00i * a11r) - (a01r * a10i + a01i * a10r);
  // b = column n of inv(A)
  float n0r = (n == 0) ? a11r : -a01r, n0i = (n == 0) ? a11i : -a01i;
  float n1r = (n == 0) ? -a10r : a00r, n1i = (n == 0) ? -a10i : a00i;
  float dmag = detr * detr + deti * deti;
  float idr = detr / dmag, idi = -deti / dmag;  // 1/det
  float b0r = n0r * idr - n0i * idi, b0i = n0r * idi + n0i * idr;
  float b1r = n1r * idr - n1i * idi, b1i = n1r * idi + n1i * idr;
  // Re(b^H D_reg b) = D00|b0|^2 + D11|b1|^2 + 2 Re(conj(b0) D01 b1)
  float cr = Dre * b1r - Dim * b1i, ci = Dre * b1i + Dim * b1r;
  float quad = D00 * (b0r * b0r + b0i * b0i) + D11 * (b1r * b1r + b1i * b1i) +
               2.f * (b0r * cr + b0i * ci);
  float den = sqrtf(quad + IP_EPS);
  w[n * 4 + 0] = b0r / den;   // w = conj(b/den)
  w[n * 4 + 1] = -b0i / den;
  w[n * 4 + 2] = b1r / den;
  w[n * 4 + 3] = -b1i / den;
}

// ---------------- final demix: out[n][j][i][c] ----------------
__global__ __launch_bounds__(256) void ilrma_final_out(
    const float* __restrict__ X, const float* __restrict__ W,
    float* __restrict__ out) {
  int idx = blockIdx.x * 256 + threadIdx.x;
  if (idx >= J_DIM * I_DIM) return;
  int j = idx / I_DIM, i = idx - j * I_DIM;
  const float* x0 = X + ((size_t)j * I_DIM + i) * 2;
  const float* x1 = X + ((size_t)(J_DIM + j) * I_DIM + i) * 2;
  float x0r = x0[0], x0i = x0[1], x1r = x1[0], x1i = x1[1];
  const float* w = W + i * 8;
#pragma unroll
  for (int nn = 0; nn < 2; ++nn) {
    float wr0 = w[nn * 4 + 0], wi0 = w[nn * 4 + 1];
    float wr1 = w[nn * 4 + 2], wi1 = w[nn * 4 + 3];
    float yr = wr0 * x0r - wi0 * x0i + wr1 * x1r - wi1 * x1i;
    float yi = wr0 * x0i + wi0 * x0r + wr1 * x1i + wi1 * x1r;
    float* o = out + ((size_t)nn * J_DIM + j) * I_DIM * 2 + (size_t)i * 2;
    o[0] = yr;
    o[1] = yi;
  }
}

// ---------------- host orchestration ----------------
extern "C" void kernel_launch(void* const* d_in, const int* in_sizes, int n_in,
                              void* d_out, int out_size, void* d_ws,
                              size_t ws_size, hipStream_t stream) {
  (void)in_sizes; (void)n_in; (void)out_size; (void)ws_size;
  const float* X = (const float*)d_in[0];
  const float* Tin = (const float*)d_in[1];
  const float* Vin = (const float*)d_in[2];
  float* out = (float*)d_out;

  size_t off = 0;
  auto take = [&](size_t nflt) -> float* {
    float* p = (float*)((char*)d_ws + off);
    off += ((nflt * sizeof(float) + 255) & ~(size_t)255);
    return p;
  };
  float* P = take((size_t)2 * J_DIM * I_DIM);
  float* Ta0 = take(I_DIM * 8); float* Ta1 = take(I_DIM * 8);
  float* Tb0 = take(I_DIM * 8); float* Tb1 = take(I_DIM * 8);
  float* V0 = take(8 * J_DIM);  float* V1 = take(8 * J_DIM);
  float* W = take(I_DIM * 8);
  float* ptN = take((size_t)JCHUNKS_T * I_DIM * 8);
  float* ptD = take((size_t)JCHUNKS_T * I_DIM * 8);
  float* pvN = take((size_t)ICHUNKS_V * 8 * J_DIM);
  float* pvD = take((size_t)ICHUNKS_V * 8 * J_DIM);

  float* Vb[2] = {V0, V1};
  float* Tcur[2] = {Ta0, Ta1};
  float* Talt[2] = {Tb0, Tb1};

  const int initBlocks = (K_DIM * J_DIM + 255) / 256;
  ilrma_init_TVW<<<initBlocks, 256, 0, stream>>>(Tin, Vin, Ta0, Ta1, V0, V1, W);

  const int pBlocks = (J_DIM * I_DIM + 255) / 256;
  const int tBlocks = (ITILES * JCHUNKS_T + 7) / 8;   // 8 waves per block
  const int tfBlocks = (I_DIM * K_DIM + 255) / 256;
  const int vBlocks = (JTILES_V * ICHUNKS_V) / 8;
  const int vfBlocks = (K_DIM * J_DIM + 255) / 256;
  const int dwBlocks = (I_DIM + 255) / 256;

  for (int it = 0; it < N_ITERS; ++it) {
    ilrma_compute_P<<<pBlocks, 256, 0, stream>>>(X, W, P);
    for (int n = 0; n < 2; ++n) {
      const float* Pn = P + (size_t)n * J_DIM * I_DIM;
      ilrma_t_update<<<tBlocks, 256, 0, stream>>>(Pn, Tcur[n], Vb[n], ptN, ptD);
      ilrma_t_finalize<<<tfBlocks, 256, 0, stream>>>(Tcur[n], ptN, ptD, Talt[n]);
      ilrma_v_accum<<<vBlocks, 256, 0, stream>>>(Pn, Tcur[n], Talt[n], Vb[n],
                                                 pvN, pvD);
      ilrma_v_finalize<<<vfBlocks, 256, 0, stream>>>(pvN, pvD, Vb[n]);
      ilrma_dw_update<<<dwBlocks, 256, 0, stream>>>(X, Talt[n], Vb[n], W, n);
      float* tmp = Tcur[n]; Tcur[n] = Talt[n]; Talt[n] = tmp;
    }
  }
  ilrma_final_out<<<pBlocks, 256, 0, stream>>>(X, W, out);
}